// TranVectorQuantizer_47777216201292
// MI455X (gfx1250) — compile-verified
//
#include <hip/hip_runtime.h>
#include <hip/hip_bf16.h>

typedef __attribute__((ext_vector_type(2))) float v2f;
typedef __attribute__((ext_vector_type(4))) float v4f;
typedef __attribute__((ext_vector_type(8))) float v8f;

// VQ argmin + gather via V_WMMA_F32_16X16X4_F32.
//   A = codebook (8x8, zero-padded to 16 rows), B = X^T for a 16-vector chunk.
//   D[r, l] = dot(codebook_r, x_l): lane l (<16) holds all 8 scores for x_l
//   in its 8 accumulator VGPRs -> lane-local argmin, no cross-lane reduce.
__global__ __launch_bounds__(256) void vq_wmma_kernel(
    const float* __restrict__ latent,
    const float* __restrict__ codebook,
    float* __restrict__ out1,   // policy_vq_latent (== quantized forward)
    float* __restrict__ out2,   // quantized_latent
    int nchunks)                // N/16 chunks of 16 vectors
{
    const int lane = threadIdx.x & 31;
    const int m    = lane & 15;   // A: codebook row | B: vector within chunk
    const int half = lane >> 4;   // selects K pair (0: K+0..1, 1: K+2..3)

    const int wave   = (blockIdx.x * blockDim.x + threadIdx.x) >> 5;
    const int nwaves = (gridDim.x * blockDim.x) >> 5;

    // --- A fragments (loop-invariant): codebook rows, zero-padded rows 8..15.
    v2f a0 = {0.f, 0.f};  // K = half*2 + {0,1}
    v2f a1 = {0.f, 0.f};  // K = 4 + half*2 + {0,1}
    if (m < 8) {
        const float* cr = codebook + m * 8 + half * 2;
        a0[0] = cr[0]; a0[1] = cr[1];
        a1[0] = cr[4]; a1[1] = cr[5];
    }

    // --- Per-lane codebook squared norms (tiny, stays in VGPRs).
    float cb2[8];
#pragma unroll
    for (int k = 0; k < 8; ++k) {
        float s = 0.f;
#pragma unroll
        for (int d = 0; d < 8; ++d) { float v = codebook[k * 8 + d]; s += v * v; }
        cb2[k] = s;
    }

    for (int ch = wave; ch < nchunks; ch += nwaves) {
        // --- B fragments: X^T for this chunk. Lane l<16 supplies x_l[K..K+1],
        //     lane l+16 supplies x_l[K+2..K+3]. 8B-aligned NT loads.
        const float* xv = latent + (size_t)ch * (16 * 8) + m * 8 + half * 2;
        v2f b0 = __builtin_nontemporal_load((const v2f*)(xv));
        v2f b1 = __builtin_nontemporal_load((const v2f*)(xv + 4));

        v8f c = {0.f, 0.f, 0.f, 0.f, 0.f, 0.f, 0.f, 0.f};
        c = __builtin_amdgcn_wmma_f32_16x16x4_f32(false, a0, false, b0,
                                                  (short)0, c, false, false);
        c = __builtin_amdgcn_wmma_f32_16x16x4_f32(false, a1, false, b1,
                                                  (short)0, c, false, false);

        // distance_r = ||c_r||^2 - 2 * dot(c_r, x)  (||x||^2 constant per row)
        int   best = 0;
        float bd   = cb2[0] - 2.f * c[0];
#pragma unroll
        for (int r = 1; r < 8; ++r) {
            float dr = cb2[r] - 2.f * c[r];
            if (dr < bd) { bd = dr; best = r; }
        }
        // Upper half-wave has zero-pad garbage; pull the winner from lane m so
        // lanes 0-15 store out1 and lanes 16-31 store out2 (no divergence).
        best = __shfl(best, m, 32);

        const v4f* q  = (const v4f*)(codebook + best * 8);
        v4f q0 = q[0], q1 = q[1];
        float* dst = (half == 0 ? out1 : out2) + ((size_t)ch * 16 + m) * 8;
        __builtin_nontemporal_store(q0, (v4f*)(dst));
        __builtin_nontemporal_store(q1, (v4f*)(dst + 4));
    }
}

// codebook_set: broadcast the 64-float codebook across the batch dimension.
__global__ __launch_bounds__(256) void vq_bcast_kernel(
    const float* __restrict__ codebook, float* __restrict__ out3, int n)
{
    int i = blockIdx.x * blockDim.x + threadIdx.x;
    if (i < n) out3[i] = codebook[i & 63];
}

extern "C" void kernel_launch(void* const* d_in, const int* in_sizes, int n_in,
                              void* d_out, int out_size, void* d_ws, size_t ws_size,
                              hipStream_t stream) {
    const float* latent   = (const float*)d_in[0];   // (4096, 1024, 8) f32
    const float* codebook = (const float*)d_in[1];   // (8, 8) f32

    const int    nelem   = in_sizes[0];              // N * 8 floats
    const int    nvec    = nelem / 8;
    const int    nchunks = nvec / 16;                // 16 vectors per WMMA chunk
    float* out1 = (float*)d_out;
    float* out2 = out1 + (size_t)nelem;
    float* out3 = out2 + (size_t)nelem;
    const int n3 = out_size - 2 * nelem;             // 4096 * 64

    // HBM-bound: 8192+ waves is plenty to saturate 23.3 TB/s.
    vq_wmma_kernel<<<2048, 256, 0, stream>>>(latent, codebook, out1, out2, nchunks);
    vq_bcast_kernel<<<(n3 + 255) / 256, 256, 0, stream>>>(codebook, out3, n3);
}